// VSSBlocktemWithDiffVivim_39848706573427
// MI455X (gfx1250) — compile-verified
//
#include <hip/hip_runtime.h>
#include <hip/hip_bf16.h>
#include <math.h>

typedef __attribute__((ext_vector_type(2))) float v2f;
typedef __attribute__((ext_vector_type(8))) float v8f;

#define BSZ   2
#define Hd    64
#define Wd    64
#define Cc    96
#define HW    4096
#define Lseq  8192
#define DI    192
#define KDIR  4
#define NST   16
#define DR    6
#define CDBL  38
#define CPAD  48
#define NCH   64      // scan chunks
#define SLEN  128     // Lseq / NCH
#define SCHUNK 32     // LDS staging depth inside a chunk
#define THETA_C   0.6f
#define EPS_LN1_C 1e-6f
#define EPS_OUT_C 1e-5f

__device__ __forceinline__ float silu_f(float v)     { return v / (1.0f + __expf(-v)); }
__device__ __forceinline__ float softplus_f(float v) { return (v > 20.0f) ? v : log1pf(__expf(v)); }

// ---------------------------------------------------------------------------
// 1. LayerNorm over C=96: one wave per row (3 elems/lane), wave-shuffle reduce
// ---------------------------------------------------------------------------
__global__ void ln_kernel(const float* __restrict__ in0, const float* __restrict__ in1,
                          const float* __restrict__ g,   const float* __restrict__ b,
                          float* __restrict__ xc, float* __restrict__ xt)
{
    int wid  = (blockIdx.x * blockDim.x + threadIdx.x) >> 5;
    int lane = threadIdx.x & 31;
    if (wid >= 2 * BSZ * HW) return;
    const float* src; float* dst; int row;
    if (wid < BSZ * HW) { src = in0; dst = xc; row = wid; }
    else                { src = in1; dst = xt; row = wid - BSZ * HW; }
    const float* p = src + (size_t)row * Cc;
    float v0 = p[lane], v1 = p[lane + 32], v2 = p[lane + 64];
    float s  = v0 + v1 + v2;
    float s2 = v0*v0 + v1*v1 + v2*v2;
    for (int off = 16; off > 0; off >>= 1) {
        s  += __shfl_xor(s,  off);
        s2 += __shfl_xor(s2, off);
    }
    float m   = s  * (1.0f / Cc);
    float var = s2 * (1.0f / Cc) - m * m;
    float r   = rsqrtf(var + EPS_LN1_C);
    float* q = dst + (size_t)row * Cc;
    q[lane]      = (v0 - m) * r * g[lane]      + b[lane];
    q[lane + 32] = (v1 - m) * r * g[lane + 32] + b[lane + 32];
    q[lane + 64] = (v2 - m) * r * g[lane + 64] + b[lane + 64];
}

// ---------------------------------------------------------------------------
// 2/7. f32 GEMM via V_WMMA_F32_16X16X4_F32: one wave per 16x16 output tile.
// ---------------------------------------------------------------------------
__global__ void gemm_wmma_f32(const float* __restrict__ A, const float* __restrict__ Bm,
                              const float* __restrict__ R, float* __restrict__ Cm,
                              int M, int N, int Kd)
{
    int ntn  = N >> 4;
    int tile = blockIdx.x;
    int tm = tile / ntn, tn = tile % ntn;
    int lane = threadIdx.x & 31;
    int half = lane >> 4, l16 = lane & 15;
    int m0 = tm << 4, n0 = tn << 4;
    v8f acc = {0.f, 0.f, 0.f, 0.f, 0.f, 0.f, 0.f, 0.f};
    const float* arow = A + (size_t)(m0 + l16) * Kd;
    for (int k0 = 0; k0 < Kd; k0 += 4) {
        int ka = k0 + 2 * half;
        __builtin_prefetch(arow + ka + 16, 0, 0);
        v2f a, b;
        a.x = arow[ka];
        a.y = arow[ka + 1];
        b.x = Bm[(size_t)ka       * N + n0 + l16];
        b.y = Bm[(size_t)(ka + 1) * N + n0 + l16];
        acc = __builtin_amdgcn_wmma_f32_16x16x4_f32(
                  false, a, false, b, (short)0, acc, false, false);
    }
#pragma unroll
    for (int r = 0; r < 8; ++r) {
        int m = m0 + r + 8 * half;
        int n = n0 + l16;
        float v = acc[r];
        if (R) v += R[(size_t)m * N + n];
        Cm[(size_t)m * N + n] = v;
    }
}

// ---------------------------------------------------------------------------
// 3. Depthwise 3x3 convs + SiLU producing x_t[b][d][t][hw]
// ---------------------------------------------------------------------------
__global__ void dwconv_kernel(const float* __restrict__ xz, const float* __restrict__ xt,
                              const float* __restrict__ w0, const float* __restrict__ b0,
                              const float* __restrict__ w1, const float* __restrict__ b1,
                              float* __restrict__ x_t)
{
    int gid = blockIdx.x * blockDim.x + threadIdx.x;
    if (gid >= BSZ * DI * HW) return;
    int hw = gid % HW;
    int d  = (gid / HW) % DI;
    int b  = gid / (HW * DI);
    int h = hw >> 6, w = hw & 63;
    float acc0 = b0[d], acc1 = b1[d];
    const float* wp0 = w0 + d * 9;
    const float* wp1 = w1 + d * 9;
    int ic = d >> 1;
    for (int kh = 0; kh < 3; kh++) {
        int hh = h + kh - 1;
        if (hh < 0 || hh >= Hd) continue;
        for (int kw = 0; kw < 3; kw++) {
            int ww = w + kw - 1;
            if (ww < 0 || ww >= Wd) continue;
            int pos = b * HW + hh * Wd + ww;
            acc0 += wp0[kh * 3 + kw] * xz[(size_t)pos * (2 * DI) + d];
            acc1 += wp1[kh * 3 + kw] * xt[(size_t)pos * Cc + ic];
        }
    }
    size_t base = (size_t)(b * DI + d) * Lseq;
    x_t[base + hw]      = silu_f(acc0);
    x_t[base + HW + hw] = silu_f(acc1);
}

// ---------------------------------------------------------------------------
// 4a. Repack x_proj weights for WMMA:
//   wpack[k][s=kh*3+kw][c(0..47)][r=i*2+t]  (c padded 38->48 with zeros)
// ---------------------------------------------------------------------------
__global__ void wpack_kernel(const float* __restrict__ xw, float* __restrict__ wpack)
{
    int gid = blockIdx.x * blockDim.x + threadIdx.x;
    if (gid >= KDIR * 9 * CPAD * 384) return;
    int r = gid % 384;
    int c = (gid / 384) % CPAD;
    int s = (gid / (384 * CPAD)) % 9;
    int k = gid / (384 * CPAD * 9);
    float v = 0.f;
    if (c < CDBL) {
        int i = r >> 1, t = r & 1;
        v = xw[(size_t)(k * CDBL + c) * DI * 18 + i * 18 + t * 9 + s];
    }
    wpack[gid] = v;
}

// 4b. kd[k][c(0..47)][i] = -THETA * sum_{t,kh,kw} W   (padded rows = 0)
__global__ void wd_kernel(const float* __restrict__ xw, float* __restrict__ wd)
{
    int gid = blockIdx.x * blockDim.x + threadIdx.x;
    if (gid >= KDIR * CPAD * DI) return;
    int i = gid % DI;
    int c = (gid / DI) % CPAD;
    int k = gid / (DI * CPAD);
    float v = 0.f;
    if (c < CDBL) {
        const float* wp = xw + (size_t)(k * CDBL + c) * DI * 18 + i * 18;
        float s = 0.f;
#pragma unroll
        for (int j = 0; j < 18; j++) s += wp[j];
        v = -THETA_C * s;
    }
    wd[gid] = v;
}

// ---------------------------------------------------------------------------
// 4c. x_proj as WMMA: out_n = sum of 9 shifted pointwise GEMMs (Kred=384),
//     then per-frame accT = out_n + (-theta*kd) @ x_t[t]  (Kred=192).
//     One wave per 16(c) x 16(hw) tile. Boundary masking is a per-lane select
//     (no EXEC divergence -> WMMA-legal).
// ---------------------------------------------------------------------------
__global__ void xproj_wmma_kernel(const float* __restrict__ x_t,
                                  const float* __restrict__ wpack,
                                  const float* __restrict__ wd,
                                  float* __restrict__ x_dbl)
{
    int id = blockIdx.x;                 // (b, k, ct, nt)
    int nt = id & 255;
    int ct = (id >> 8) % 3;
    int k  = (id / 768) % KDIR;
    int b  = id / (768 * KDIR);
    int lane = threadIdx.x & 31;
    int half = lane >> 4, l16 = lane & 15;
    int n0 = nt << 4, c0 = ct << 4;
    int hw = n0 + l16;
    int h = hw >> 6, w = hw & 63;
    const float* xbase = x_t + (size_t)b * (2 * DI) * HW;   // rows r=i*2+t, stride HW

    v8f accN = {0.f, 0.f, 0.f, 0.f, 0.f, 0.f, 0.f, 0.f};
    for (int s = 0; s < 9; s++) {
        int dh = s / 3 - 1, dw = s % 3 - 1;
        int hh = h + dh, ww = w + dw;
        bool valid = (hh >= 0 && hh < Hd && ww >= 0 && ww < Wd);
        int hwp = hh * Wd + ww;
        const float* arow = wpack + (size_t)((k * 9 + s) * CPAD + c0 + l16) * 384;
        for (int k0 = 0; k0 < 384; k0 += 4) {
            int ka = k0 + 2 * half;
            __builtin_prefetch(arow + ka + 16, 0, 0);
            v2f a, bv;
            a.x = arow[ka];
            a.y = arow[ka + 1];
            bv.x = valid ? xbase[(size_t)ka       * HW + hwp] : 0.f;
            bv.y = valid ? xbase[(size_t)(ka + 1) * HW + hwp] : 0.f;
            accN = __builtin_amdgcn_wmma_f32_16x16x4_f32(
                       false, a, false, bv, (short)0, accN, false, false);
        }
    }
    const float* kdrow = wd + (size_t)(k * CPAD + c0 + l16) * DI;
    v8f accT[2];
#pragma unroll
    for (int t = 0; t < 2; t++) {
        v8f acc = accN;
        for (int k0 = 0; k0 < DI; k0 += 4) {
            int ka = k0 + 2 * half;
            v2f a, bv;
            a.x = kdrow[ka];
            a.y = kdrow[ka + 1];
            bv.x = xbase[(size_t)(2 * ka + t)     * HW + hw];
            bv.y = xbase[(size_t)(2 * ka + 2 + t) * HW + hw];
            acc = __builtin_amdgcn_wmma_f32_16x16x4_f32(
                      false, a, false, bv, (short)0, acc, false, false);
        }
        accT[t] = acc;
    }
    size_t obase = (size_t)(b * KDIR + k) * CDBL * Lseq;
#pragma unroll
    for (int t = 0; t < 2; t++) {
#pragma unroll
        for (int r = 0; r < 8; r++) {
            int c = c0 + r + 8 * half;
            if (c < CDBL)
                x_dbl[obase + (size_t)c * Lseq + t * HW + hw] = accT[t][r];
        }
    }
}

// ---------------------------------------------------------------------------
// 5. Blocked parallel scan (3 passes).
//    Recurrence: h <- exp(dt*A)*h + (dt*u)*B ; chunk decay product = exp(A*sum dt).
//    Pass A (WRITE_Y=false): per-chunk local solve from h=0 -> F, sum(dt).
//    Pass B: sequential 64-chunk prefix per (b,k,d) -> Hstart.
//    Pass C (WRITE_Y=true): re-run chunk from Hstart, emit y.
// ---------------------------------------------------------------------------
template <bool WRITE_Y>
__global__ void scan_chunk_kernel(const float* __restrict__ x_dbl, const float* __restrict__ x_t,
                                  const float* __restrict__ dt_w,  const float* __restrict__ dt_b,
                                  const float* __restrict__ A_logs,const float* __restrict__ Ds,
                                  const float* __restrict__ Hstart,
                                  float* __restrict__ Fout, float* __restrict__ Sdtout,
                                  float* __restrict__ out_y)
{
    int blk = blockIdx.x;                // (b*KDIR + k) * NCH + chunk
    int ch  = blk % NCH;
    int bk  = blk / NCH;
    int b = bk / KDIR, k = bk % KDIR;
    int d = threadIdx.x;                 // 0..191
    float Ar[NST];
#pragma unroll
    for (int n = 0; n < NST; n++) Ar[n] = -__expf(A_logs[(size_t)(k * DI + d) * NST + n]);
    float dtw[DR];
#pragma unroll
    for (int r = 0; r < DR; r++) dtw[r] = dt_w[(size_t)(k * DI + d) * DR + r];
    float dtb = dt_b[k * DI + d];
    float Dd  = Ds[k * DI + d];
    size_t scan_idx = (size_t)(bk * DI + d) * NCH + ch;

    float hst[NST];
    if (WRITE_Y) {
#pragma unroll
        for (int n = 0; n < NST; n++) hst[n] = Hstart[scan_idx * NST + n];
    } else {
#pragma unroll
        for (int n = 0; n < NST; n++) hst[n] = 0.f;
    }
    float sdt = 0.f;

    __shared__ float sdtL[SCHUNK][DR];
    __shared__ float sB  [SCHUNK][NST];
    __shared__ float sC  [SCHUNK][NST];

    const float* dbase = x_dbl + (size_t)bk * CDBL * Lseq;
    const float* ubase = x_t   + (size_t)(b * DI + d) * Lseq;
    float*       obase = out_y + (size_t)(bk * DI + d) * Lseq;
    bool rev = (k & 1);

    int lbeg = ch * SLEN;
    for (int l0 = lbeg; l0 < lbeg + SLEN; l0 += SCHUNK) {
        __syncthreads();
        for (int idx = threadIdx.x; idx < CDBL * SCHUNK; idx += blockDim.x) {
            int row = idx / SCHUNK, j = idx % SCHUNK;
            float v = dbase[(size_t)row * Lseq + l0 + j];
            if      (row < DR)       sdtL[j][row] = v;
            else if (row < DR + NST) sB[j][row - DR] = v;
            else                     sC[j][row - DR - NST] = v;
        }
        __syncthreads();
        for (int j = 0; j < SCHUNK; j++) {
            int l = l0 + j;
            float dt = dtb;
#pragma unroll
            for (int r = 0; r < DR; r++) dt += sdtL[j][r] * dtw[r];
            dt = softplus_f(dt);
            sdt += dt;
            float u  = ubase[rev ? (Lseq - 1 - l) : l];
            float du = dt * u;
            float y  = 0.f;
#pragma unroll
            for (int n = 0; n < NST; n++) {
                float dA = __expf(dt * Ar[n]);
                hst[n] = dA * hst[n] + du * sB[j][n];
                y += hst[n] * sC[j][n];
            }
            if (WRITE_Y) obase[l] = y + u * Dd;
        }
    }
    if (!WRITE_Y) {
#pragma unroll
        for (int n = 0; n < NST; n++) Fout[scan_idx * NST + n] = hst[n];
        Sdtout[scan_idx] = sdt;
    }
}

// Pass B: one thread per (b,k,d); prefix over NCH chunks.
__global__ void scan_prefix_kernel(const float* __restrict__ F, const float* __restrict__ Sdt,
                                   const float* __restrict__ A_logs,
                                   float* __restrict__ Hstart)
{
    int bkd = blockIdx.x * blockDim.x + threadIdx.x;
    if (bkd >= BSZ * KDIR * DI) return;
    int d  = bkd % DI;
    int bk = bkd / DI;
    int k  = bk % KDIR;
    float Ar[NST];
#pragma unroll
    for (int n = 0; n < NST; n++) Ar[n] = -__expf(A_logs[(size_t)(k * DI + d) * NST + n]);
    float h[NST];
#pragma unroll
    for (int n = 0; n < NST; n++) h[n] = 0.f;
    for (int c = 0; c < NCH; c++) {
        size_t idx = (size_t)bkd * NCH + c;
        float sdt = Sdt[idx];
#pragma unroll
        for (int n = 0; n < NST; n++) {
            Hstart[idx * NST + n] = h[n];
            float p = __expf(sdt * Ar[n]);
            h[n] = p * h[n] + F[idx * NST + n];
        }
    }
}

// ---------------------------------------------------------------------------
// 6. Merge 4 directions (+reversals, +HW transposes), LayerNorm(192), SiLU gate
// ---------------------------------------------------------------------------
__global__ void merge_kernel(const float* __restrict__ out_y, const float* __restrict__ xz,
                             const float* __restrict__ g,     const float* __restrict__ bb,
                             float* __restrict__ yg)
{
    int wid  = (blockIdx.x * blockDim.x + threadIdx.x) >> 5;
    int lane = threadIdx.x & 31;
    if (wid >= BSZ * HW) return;
    int b = wid / HW, hw = wid % HW;
    int h = hw >> 6, w = hw & 63;
    int lt = w * 64 + h;
    float acc[6];
    float s = 0.f, s2 = 0.f;
#pragma unroll
    for (int j = 0; j < 6; j++) {
        int d = lane + 32 * j;
        size_t b0 = (size_t)((b * KDIR + 0) * DI + d) * Lseq;
        size_t b1 = (size_t)((b * KDIR + 1) * DI + d) * Lseq;
        size_t b2 = (size_t)((b * KDIR + 2) * DI + d) * Lseq;
        size_t b3 = (size_t)((b * KDIR + 3) * DI + d) * Lseq;
        float a = out_y[b0 + hw]              + out_y[b0 + HW + hw]
                + out_y[b2 + (Lseq - 1 - hw)] + out_y[b2 + (Lseq - 1 - (HW + hw))]
                + out_y[b1 + lt]              + out_y[b1 + HW + lt]
                + out_y[b3 + (Lseq - 1 - lt)] + out_y[b3 + (Lseq - 1 - (HW + lt))];
        acc[j] = a;
        s += a; s2 += a * a;
    }
    for (int off = 16; off > 0; off >>= 1) {
        s  += __shfl_xor(s,  off);
        s2 += __shfl_xor(s2, off);
    }
    float m   = s  * (1.0f / DI);
    float var = s2 * (1.0f / DI) - m * m;
    float r   = rsqrtf(var + EPS_OUT_C);
    size_t rowz = (size_t)(b * HW + hw) * (2 * DI);
    size_t rowo = (size_t)(b * HW + hw) * DI;
#pragma unroll
    for (int j = 0; j < 6; j++) {
        int d = lane + 32 * j;
        float yn = (acc[j] - m) * r * g[d] + bb[d];
        float z  = xz[rowz + DI + d];
        yg[rowo + d] = yn * silu_f(z);
    }
}

// ---------------------------------------------------------------------------
extern "C" void kernel_launch(void* const* d_in, const int* in_sizes, int n_in,
                              void* d_out, int out_size, void* d_ws, size_t ws_size,
                              hipStream_t stream)
{
    (void)in_sizes; (void)n_in; (void)out_size; (void)ws_size;
    const float* input0     = (const float*)d_in[0];
    const float* input1     = (const float*)d_in[1];
    const float* ln1_g      = (const float*)d_in[2];
    const float* ln1_b      = (const float*)d_in[3];
    const float* in_proj_w  = (const float*)d_in[4];
    const float* conv2d_w   = (const float*)d_in[5];
    const float* conv2d_b   = (const float*)d_in[6];
    const float* conv2dxt_w = (const float*)d_in[7];
    const float* conv2dxt_b = (const float*)d_in[8];
    const float* x_proj_w   = (const float*)d_in[9];
    const float* dt_w       = (const float*)d_in[10];
    const float* dt_b       = (const float*)d_in[11];
    const float* A_logs     = (const float*)d_in[12];
    const float* Ds         = (const float*)d_in[13];
    const float* out_norm_g = (const float*)d_in[14];
    const float* out_norm_b = (const float*)d_in[15];
    const float* out_proj_w = (const float*)d_in[16];

    float* out0   = (float*)d_out;
    float* xt_out = out0 + (size_t)BSZ * HW * Cc;     // output1 == LN(input1)

    float* ws = (float*)d_ws;
    size_t off = 0;
    float* xc    = ws + off; off += (size_t)BSZ * HW * Cc;            // 786432
    float* xz    = ws + off; off += (size_t)BSZ * HW * 2 * DI;        // 3145728
    float* x_t   = ws + off; off += (size_t)BSZ * DI * Lseq;          // 3145728
    float* x_dbl = ws + off; off += (size_t)BSZ * KDIR * CDBL * Lseq; // 2490368
    float* outy  = ws + off; off += (size_t)BSZ * KDIR * DI * Lseq;   // 12582912
    float* yg    = ws + off; off += (size_t)BSZ * HW * DI;            // 1572864
    float* wpack = ws + off; off += (size_t)KDIR * 9 * CPAD * 384;    // 663552
    float* wdbuf = ws + off; off += (size_t)KDIR * CPAD * DI;         // 36864
    float* Fbuf  = ws + off; off += (size_t)BSZ * KDIR * DI * NCH * NST; // 1572864
    float* Sdtb  = ws + off; off += (size_t)BSZ * KDIR * DI * NCH;    // 98304
    float* Hst   = ws + off;                                          // 1572864

    // weight repack (independent of activations; launch first)
    {
        int total = KDIR * 9 * CPAD * 384;
        wpack_kernel<<<(total + 255) / 256, 256, 0, stream>>>(x_proj_w, wpack);
        total = KDIR * CPAD * DI;
        wd_kernel<<<(total + 255) / 256, 256, 0, stream>>>(x_proj_w, wdbuf);
    }
    // 1. LayerNorm both inputs (xt goes straight into d_out slot 1)
    {
        int waves = 2 * BSZ * HW;
        ln_kernel<<<(waves * 32 + 255) / 256, 256, 0, stream>>>(input0, input1, ln1_g, ln1_b, xc, xt_out);
    }
    // 2. in_proj GEMM (WMMA f32): [8192x96] x [96x384]
    {
        int M = BSZ * HW, N = 2 * DI, Kd = Cc;
        gemm_wmma_f32<<<(M / 16) * (N / 16), 32, 0, stream>>>(xc, in_proj_w, nullptr, xz, M, N, Kd);
    }
    // 3. depthwise convs + SiLU -> x_t
    {
        int total = BSZ * DI * HW;
        dwconv_kernel<<<(total + 255) / 256, 256, 0, stream>>>(
            xz, xt_out, conv2d_w, conv2d_b, conv2dxt_w, conv2dxt_b, x_t);
    }
    // 4. x_proj as 9 shifted WMMA GEMMs + pointwise correction -> x_dbl
    xproj_wmma_kernel<<<BSZ * KDIR * 3 * 256, 32, 0, stream>>>(x_t, wpack, wdbuf, x_dbl);
    // 5. blocked parallel selective scan
    scan_chunk_kernel<false><<<BSZ * KDIR * NCH, DI, 0, stream>>>(
        x_dbl, x_t, dt_w, dt_b, A_logs, Ds, nullptr, Fbuf, Sdtb, nullptr);
    {
        int total = BSZ * KDIR * DI;
        scan_prefix_kernel<<<(total + 127) / 128, 128, 0, stream>>>(Fbuf, Sdtb, A_logs, Hst);
    }
    scan_chunk_kernel<true><<<BSZ * KDIR * NCH, DI, 0, stream>>>(
        x_dbl, x_t, dt_w, dt_b, A_logs, Ds, Hst, nullptr, nullptr, outy);
    // 6. merge directions + out LayerNorm + SiLU gate
    {
        int waves = BSZ * HW;
        merge_kernel<<<(waves * 32 + 255) / 256, 256, 0, stream>>>(outy, xz, out_norm_g, out_norm_b, yg);
    }
    // 7. out_proj GEMM (WMMA f32) + residual xc -> output0
    {
        int M = BSZ * HW, N = Cc, Kd = DI;
        gemm_wmma_f32<<<(M / 16) * (N / 16), 32, 0, stream>>>(yg, out_proj_w, xc, out0, M, N, Kd);
    }
}